// PointNetSetAbstraction_39213051412827
// MI455X (gfx1250) — compile-verified
//
#include <hip/hip_runtime.h>
#include <cfloat>

typedef float v2f __attribute__((ext_vector_type(2)));
typedef float v8f __attribute__((ext_vector_type(8)));

constexpr int   BATCH = 4;
constexpr int   NPTS  = 16384;
constexpr int   CIN_  = 32;
constexpr int   DIM_  = 256;
constexpr int   NS    = 2048;   // npoint
constexpr int   NK    = 16;     // nneigh
constexpr float BNEPS = 1e-5f;
constexpr int   ROWS  = BATCH * NPTS;   // 65536

// ---------------------------------------------------------------------------
// Strip GEMM: each wave computes a 16-row x 128-col strip of C = A*B + bias
// as 8 WMMA tiles sharing one A fetch per k-step (V_WMMA_F32_16X16X4_F32).
// 8 x v8f accumulators = 64 VGPRs -> no spills; B loads for a k-step are
// issued as a clause before the 8 WMMAs so waits stagger instead of
// serializing. launch_bounds(128,1) relaxes the VGPR cap.
// TRANSB=false: B[k][n] = Bm[k*DIM + n]    (fc1 weight [CIN, DIM])
// TRANSB=true : B[k][n] = Bm[n*KDIM + k]   (conv weight [DIM,DIM], h @ W^T)
// ---------------------------------------------------------------------------
template<int KDIM, bool TRANSB>
__global__ __launch_bounds__(128, 1)
void gemm_wmma_strip(const float* __restrict__ A,
                     const float* __restrict__ Bm,
                     const float* __restrict__ bias,
                     float* __restrict__ C,
                     int M) {
  constexpr int NT   = 8;                 // N-tiles per wave (128 cols)
  constexpr int NGRP = DIM_ / (16 * NT);  // 2 column groups
  const int lane = threadIdx.x & 31;
  const int wave = blockIdx.x * (blockDim.x >> 5) + (threadIdx.x >> 5);
  const int tM = wave / NGRP;
  const int tg = wave % NGRP;             // which 128-col group
  if (tM * 16 >= M) return;

  const int m    = lane & 15;   // A row / B col / C col within tile
  const int half = lane >> 4;   // K-pair selector (A/B), M-half selector (C)
  const int col0 = tg * NT * 16;          // first column of this strip

  v8f acc[NT];
#pragma unroll
  for (int t = 0; t < NT; ++t) {
    const float bv = bias[col0 + t * 16 + m];
#pragma unroll
    for (int r = 0; r < 8; ++r) acc[t][r] = bv;
  }

  const float* arow  = A + (size_t)(tM * 16 + m) * KDIM + 2 * half;
  const float* bbase = TRANSB ? (Bm + (size_t)(col0 + m) * KDIM + 2 * half)
                              : (Bm + (size_t)(2 * half) * DIM_ + col0 + m);

  for (int kk = 0; kk < KDIM; kk += 4) {
    const v2f a = *(const v2f*)(arow + kk);   // shared by all 8 tiles

    v2f breg[NT];                             // clause the 8 B loads together
#pragma unroll
    for (int t = 0; t < NT; ++t) {
      if (TRANSB) {
        breg[t] = *(const v2f*)(bbase + (size_t)t * 16 * KDIM + kk);
      } else {
        breg[t].x = bbase[(size_t)kk * DIM_ + t * 16];
        breg[t].y = bbase[(size_t)(kk + 1) * DIM_ + t * 16];
      }
    }
#pragma unroll
    for (int t = 0; t < NT; ++t)
      acc[t] = __builtin_amdgcn_wmma_f32_16x16x4_f32(false, a, false, breg[t],
                                                     (short)0, acc[t],
                                                     false, false);
  }

  // C/D layout: VGPR r -> M = r + 8*half, lane&15 -> N; single base + imm offsets
  float* cbase = C + (size_t)(tM * 16 + 8 * half) * DIM_ + col0 + m;
#pragma unroll
  for (int r = 0; r < 8; ++r)
#pragma unroll
    for (int t = 0; t < NT; ++t)
      cbase[(size_t)r * DIM_ + t * 16] = acc[t][r];
}

// ---------------------------------------------------------------------------
// BatchNorm (training forward): stats over all rows, per channel (DIM=256).
// ---------------------------------------------------------------------------
__global__ void zero_stats_kernel(float* stats) {
  stats[threadIdx.x] = 0.0f;           // 512 entries: sum[256], sumsq[256]
}

__global__ void bn_stats_kernel(const float* __restrict__ X,
                                float* __restrict__ sum,
                                float* __restrict__ sumsq,
                                int R) {
  const int c   = threadIdx.x;          // channel
  const int rpb = R / gridDim.x;        // rows per block (exact divide)
  const int r0  = blockIdx.x * rpb;
  float s = 0.0f, q = 0.0f;
  for (int r = r0; r < r0 + rpb; ++r) {
    float v = X[(size_t)r * DIM_ + c];
    s += v;
    q += v * v;
  }
  atomicAdd(sum + c, s);
  atomicAdd(sumsq + c, q);
}

__global__ void bn_finalize_kernel(const float* __restrict__ sum,
                                   const float* __restrict__ sumsq,
                                   const float* __restrict__ g,
                                   const float* __restrict__ b,
                                   float* __restrict__ scale,
                                   float* __restrict__ shift,
                                   float invR) {
  const int c = threadIdx.x;
  float mean = sum[c] * invR;
  float var  = sumsq[c] * invR - mean * mean;   // biased variance
  float sc   = g[c] * rsqrtf(var + BNEPS);
  scale[c] = sc;
  shift[c] = b[c] - mean * sc;
}

__global__ void bnrelu_inplace_kernel(float* __restrict__ X,
                                      const float* __restrict__ scale,
                                      const float* __restrict__ shift,
                                      int count) {
  int i = blockIdx.x * blockDim.x + threadIdx.x;
  if (i >= count) return;
  int c = i & (DIM_ - 1);
  X[i] = fmaxf(0.0f, X[i] * scale[c] + shift[c]);
}

// h1 = h0 + relu(bn2(u))  (in place on u)
__global__ void residual_bnrelu_kernel(float* __restrict__ U,
                                       const float* __restrict__ H0,
                                       const float* __restrict__ scale,
                                       const float* __restrict__ shift,
                                       int count) {
  int i = blockIdx.x * blockDim.x + threadIdx.x;
  if (i >= count) return;
  int c = i & (DIM_ - 1);
  U[i] = H0[i] + fmaxf(0.0f, U[i] * scale[c] + shift[c]);
}

__global__ void bn_apply_out_kernel(const float* __restrict__ P,
                                    const float* __restrict__ scale,
                                    const float* __restrict__ shift,
                                    float* __restrict__ out,
                                    int count) {
  int i = blockIdx.x * blockDim.x + threadIdx.x;
  if (i >= count) return;
  int c = i & (DIM_ - 1);
  out[i] = P[i] * scale[c] + shift[c];
}

// ---------------------------------------------------------------------------
// Furthest point sampling: one 1024-thread block per batch, points and running
// min-distance held in registers (16 per thread), argmax via LDS tree.
// Also emits new_xyz = gather(xyz, fps_idx) directly into d_out.
// ---------------------------------------------------------------------------
__global__ __launch_bounds__(1024)
void fps_kernel(const float* __restrict__ xyz,
                int* __restrict__ fps_idx,
                float* __restrict__ out_xyz) {
  constexpr int PPT = NPTS / 1024;      // 16 points per thread
  const int b   = blockIdx.x;
  const int tid = threadIdx.x;
  const float* xb = xyz + (size_t)b * NPTS * 3;

  float px[PPT], py[PPT], pz[PPT], dist[PPT];
#pragma unroll
  for (int i = 0; i < PPT; ++i) {
    int p = tid + i * 1024;
    px[i] = xb[p * 3 + 0];
    py[i] = xb[p * 3 + 1];
    pz[i] = xb[p * 3 + 2];
    dist[i] = FLT_MAX;
  }

  __shared__ float s_val[1024];
  __shared__ int   s_idx[1024];
  __shared__ float cent[3];

  int far = 0;  // only meaningful on tid==0
  for (int s = 0; s < NS; ++s) {
    if (tid == 0) {
      fps_idx[b * NS + s] = far;
      float cx = xb[far * 3 + 0], cy = xb[far * 3 + 1], cz = xb[far * 3 + 2];
      cent[0] = cx; cent[1] = cy; cent[2] = cz;
      float* o = out_xyz + (size_t)(b * NS + s) * 3;
      o[0] = cx; o[1] = cy; o[2] = cz;
    }
    __syncthreads();
    const float cx = cent[0], cy = cent[1], cz = cent[2];

    float best = -1.0f; int bidx = 0x7fffffff;
#pragma unroll
    for (int i = 0; i < PPT; ++i) {
      float dx = px[i] - cx, dy = py[i] - cy, dz = pz[i] - cz;
      float d = dx * dx + dy * dy + dz * dz;
      dist[i] = fminf(dist[i], d);
      int p = tid + i * 1024;
      if (dist[i] > best || (dist[i] == best && p < bidx)) { best = dist[i]; bidx = p; }
    }
    s_val[tid] = best; s_idx[tid] = bidx;
    __syncthreads();
    for (int off = 512; off > 0; off >>= 1) {
      if (tid < off) {
        float v2 = s_val[tid + off]; int i2 = s_idx[tid + off];
        if (v2 > s_val[tid] || (v2 == s_val[tid] && i2 < s_idx[tid])) {
          s_val[tid] = v2; s_idx[tid] = i2;
        }
      }
      __syncthreads();
    }
    if (tid == 0) far = s_idx[0];
    __syncthreads();
  }
}

// ---------------------------------------------------------------------------
// kNN (top-16 smallest sq-dist of 16384) + gather/max + points_ori residual.
// One 256-thread block per query; d[] in 64KB dynamic LDS (xyz sits in L2).
// np_pre[b,s,c] = h0[fps_idx[b,s]][c] + max_k h1[knn_k][c]
// ---------------------------------------------------------------------------
__global__ void knn_group_kernel(const float* __restrict__ xyz,
                                 const float* __restrict__ new_xyz,
                                 const float* __restrict__ h0,
                                 const float* __restrict__ h1,
                                 const int* __restrict__ fps_idx,
                                 float* __restrict__ np_pre) {
  extern __shared__ float d[];          // NPTS floats
  __shared__ float rv[256];
  __shared__ int   ri[256];
  __shared__ int   sel[NK];

  const int b   = blockIdx.x / NS;
  const int s   = blockIdx.x % NS;
  const int tid = threadIdx.x;

  const float* q = new_xyz + (size_t)(b * NS + s) * 3;
  const float qx = q[0], qy = q[1], qz = q[2];
  const float* xb = xyz + (size_t)b * NPTS * 3;

  for (int p = tid; p < NPTS; p += 256) {
    float dx = xb[p * 3 + 0] - qx;
    float dy = xb[p * 3 + 1] - qy;
    float dz = xb[p * 3 + 2] - qz;
    d[p] = dx * dx + dy * dy + dz * dz;
  }
  __syncthreads();

  for (int k = 0; k < NK; ++k) {
    float best = FLT_MAX; int bi = 0x7fffffff;
    for (int p = tid; p < NPTS; p += 256) {
      float v = d[p];
      if (v < best || (v == best && p < bi)) { best = v; bi = p; }
    }
    rv[tid] = best; ri[tid] = bi;
    __syncthreads();
    for (int off = 128; off > 0; off >>= 1) {
      if (tid < off) {
        float v2 = rv[tid + off]; int i2 = ri[tid + off];
        if (v2 < rv[tid] || (v2 == rv[tid] && i2 < ri[tid])) {
          rv[tid] = v2; ri[tid] = i2;
        }
      }
      __syncthreads();
    }
    if (tid == 0) { sel[k] = ri[0]; d[ri[0]] = FLT_MAX; }
    __syncthreads();
  }

  // gather + max over neighbors, fused with points_ori residual
  const int c = tid;                    // 256 threads == DIM channels
  const float* h1b = h1 + (size_t)b * NPTS * DIM_;
  float mx = -FLT_MAX;
#pragma unroll
  for (int k = 0; k < NK; ++k)
    mx = fmaxf(mx, h1b[(size_t)sel[k] * DIM_ + c]);
  const int fi = fps_idx[b * NS + s];
  np_pre[(size_t)(b * NS + s) * DIM_ + c] =
      h0[((size_t)b * NPTS + fi) * DIM_ + c] + mx;
}

// ---------------------------------------------------------------------------
// Host-side orchestration
// ---------------------------------------------------------------------------
extern "C" void kernel_launch(void* const* d_in, const int* in_sizes, int n_in,
                              void* d_out, int out_size, void* d_ws, size_t ws_size,
                              hipStream_t stream) {
  (void)in_sizes; (void)n_in; (void)out_size; (void)ws_size;
  const float* xyz     = (const float*)d_in[0];
  const float* points  = (const float*)d_in[1];
  const float* fc1_w   = (const float*)d_in[2];
  const float* fc1_b   = (const float*)d_in[3];
  const float* conv1_w = (const float*)d_in[4];
  const float* conv1_b = (const float*)d_in[5];
  const float* conv2_w = (const float*)d_in[6];
  const float* conv2_b = (const float*)d_in[7];
  const float* bn1_g   = (const float*)d_in[8];
  const float* bn1_b   = (const float*)d_in[9];
  const float* bn2_g   = (const float*)d_in[10];
  const float* bn2_b   = (const float*)d_in[11];
  const float* bn_g    = (const float*)d_in[12];
  const float* bn_b    = (const float*)d_in[13];

  // workspace layout (bytes)
  char* ws = (char*)d_ws;
  constexpr size_t BIG = (size_t)ROWS * DIM_ * sizeof(float);   // 64 MB
  float* h0    = (float*)(ws);                    // fc1 output
  float* T     = (float*)(ws + BIG);              // conv1 pre/post BN
  float* U     = (float*)(ws + 2 * BIG);          // conv2 pre -> h1
  float* Pnp   = (float*)(ws + 3 * BIG);          // np_pre [B*S, DIM], 8 MB
  float* stats = (float*)(ws + 3 * BIG + (size_t)BATCH * NS * DIM_ * sizeof(float));
  float* sum   = stats;          // 256
  float* sumsq = stats + 256;    // 256
  float* scale = stats + 512;    // 256
  float* shift = stats + 768;    // 256
  int*   fpsI  = (int*)(stats + 1024);            // B*NS ints

  float* out_xyz = (float*)d_out;                       // [B,S,3]
  float* out_pts = (float*)d_out + (size_t)BATCH * NS * 3;

  const int ELT  = ROWS * DIM_;                   // 16,777,216
  const int SELT = BATCH * NS * DIM_;             // 2,097,152
  // 1 wave per 16x128 strip, 2 strips per row-block, 4 waves/block
  const int strip_blocks = (ROWS / 16) * 2 / 4;   // 2048 blocks

  // FPS (independent of GEMM chain) -> fps_idx + new_xyz
  fps_kernel<<<BATCH, 1024, 0, stream>>>(xyz, fpsI, out_xyz);

  // h0 = points @ fc1_w + fc1_b
  gemm_wmma_strip<CIN_, false><<<strip_blocks, 128, 0, stream>>>(
      points, fc1_w, fc1_b, h0, ROWS);

  // t_pre = h0 @ conv1_w^T + conv1_b
  gemm_wmma_strip<DIM_, true><<<strip_blocks, 128, 0, stream>>>(
      h0, conv1_w, conv1_b, T, ROWS);

  // bn1 + relu (in place on T)
  zero_stats_kernel<<<1, 512, 0, stream>>>(stats);
  bn_stats_kernel<<<256, 256, 0, stream>>>(T, sum, sumsq, ROWS);
  bn_finalize_kernel<<<1, 256, 0, stream>>>(sum, sumsq, bn1_g, bn1_b,
                                            scale, shift, 1.0f / ROWS);
  bnrelu_inplace_kernel<<<ELT / 256, 256, 0, stream>>>(T, scale, shift, ELT);

  // u_pre = T @ conv2_w^T + conv2_b
  gemm_wmma_strip<DIM_, true><<<strip_blocks, 128, 0, stream>>>(
      T, conv2_w, conv2_b, U, ROWS);

  // h1 = h0 + relu(bn2(u_pre))  (in place on U)
  zero_stats_kernel<<<1, 512, 0, stream>>>(stats);
  bn_stats_kernel<<<256, 256, 0, stream>>>(U, sum, sumsq, ROWS);
  bn_finalize_kernel<<<1, 256, 0, stream>>>(sum, sumsq, bn2_g, bn2_b,
                                            scale, shift, 1.0f / ROWS);
  residual_bnrelu_kernel<<<ELT / 256, 256, 0, stream>>>(U, h0, scale, shift, ELT);

  // kNN + gather/max + points_ori residual -> np_pre
  knn_group_kernel<<<BATCH * NS, 256, NPTS * sizeof(float), stream>>>(
      xyz, out_xyz, h0, U, fpsI, Pnp);

  // final bn -> output
  zero_stats_kernel<<<1, 512, 0, stream>>>(stats);
  bn_stats_kernel<<<32, 256, 0, stream>>>(Pnp, sum, sumsq, BATCH * NS);
  bn_finalize_kernel<<<1, 256, 0, stream>>>(sum, sumsq, bn_g, bn_b,
                                            scale, shift, 1.0f / (BATCH * NS));
  bn_apply_out_kernel<<<SELT / 256, 256, 0, stream>>>(Pnp, scale, shift,
                                                      out_pts, SELT);
}